// PointNetPlusPlusSegmentation_16758962389306
// MI455X (gfx1250) — compile-verified
//
#include <hip/hip_runtime.h>
#include <hip/hip_bf16.h>

typedef __attribute__((ext_vector_type(2))) float v2f;
typedef __attribute__((ext_vector_type(8))) float v8f;

// ---------------------------------------------------------------------------
// Utility: deterministic hash -> pseudo-random index (stands in for jax PRNG)
// ---------------------------------------------------------------------------
__device__ __forceinline__ unsigned hash3(unsigned a, unsigned b, unsigned c) {
  unsigned h = a * 0x9E3779B1u;
  h ^= b + 0x85EBCA77u + (h << 6) + (h >> 2);
  h ^= c + 0xC2B2AE3Du + (h << 6) + (h >> 2);
  h ^= h >> 16; h *= 0x7FEB352Du;
  h ^= h >> 15; h *= 0x846CA68Bu;
  h ^= h >> 16;
  return h;
}

// ---------------------------------------------------------------------------
// WMMA GEMM: Y[Opad x P] = Wp[Opad x Cpad] * X[Cpad x P]   (fp32, 16x16x4)
// Each wave computes FOUR 16x16 M-tiles (64 output channels) x one N-tile,
// reusing the B fragment 4x per K-step (4x less X-operand traffic).
// 8 waves per 256-thread block cover a 128-point N stripe.
// A frag (16x4): lane m=lane&15; VGPR0/1 hold K = (lane<16 ? 0,1 : 2,3)
// B frag (4x16): lane n=lane&15; VGPR0/1 hold K = (lane<16 ? 0,1 : 2,3)
// C/D (16x16):  lane n=lane&15; VGPR r holds M = r + (lane<16 ? 0 : 8)
// ---------------------------------------------------------------------------
__global__ __launch_bounds__(256) void wmma_gemm_f32(
    const float* __restrict__ Wp, const float* __restrict__ X,
    float* __restrict__ Y, int Cpad, int P) {
  int wave = threadIdx.x >> 5;
  int lane = threadIdx.x & 31;
  int tileN = blockIdx.x * 8 + wave;
  int tileM4 = blockIdx.y;               // group of 4 M-tiles (64 channels)
  int n = tileN * 16 + (lane & 15);
  int kh = (lane >> 4) << 1;             // 0 or 2
  const float* wptr = Wp + (size_t)(tileM4 * 64 + (lane & 15)) * Cpad + kh;
  const float* xptr = X + (size_t)kh * P + n;
  v8f acc0 = {}, acc1 = {}, acc2 = {}, acc3 = {};
  for (int k = 0; k < Cpad; k += 4) {
    v2f b;
    b.x = xptr[(size_t)k * P];
    b.y = xptr[(size_t)(k + 1) * P];
    __builtin_prefetch(&xptr[(size_t)(k + 4) * P], 0, 1);
    v2f a0, a1, a2, a3;
    a0.x = wptr[(size_t)0 * 16 * Cpad + k];
    a0.y = wptr[(size_t)0 * 16 * Cpad + k + 1];
    a1.x = wptr[(size_t)1 * 16 * Cpad + k];
    a1.y = wptr[(size_t)1 * 16 * Cpad + k + 1];
    a2.x = wptr[(size_t)2 * 16 * Cpad + k];
    a2.y = wptr[(size_t)2 * 16 * Cpad + k + 1];
    a3.x = wptr[(size_t)3 * 16 * Cpad + k];
    a3.y = wptr[(size_t)3 * 16 * Cpad + k + 1];
    acc0 = __builtin_amdgcn_wmma_f32_16x16x4_f32(false, a0, false, b,
                                                 (short)0, acc0, false, false);
    acc1 = __builtin_amdgcn_wmma_f32_16x16x4_f32(false, a1, false, b,
                                                 (short)0, acc1, false, false);
    acc2 = __builtin_amdgcn_wmma_f32_16x16x4_f32(false, a2, false, b,
                                                 (short)0, acc2, false, false);
    acc3 = __builtin_amdgcn_wmma_f32_16x16x4_f32(false, a3, false, b,
                                                 (short)0, acc3, false, false);
  }
  int mhalf = (lane >> 4) << 3;          // 0 or 8
  size_t ybase = (size_t)(tileM4 * 64 + mhalf) * P + n;
#pragma unroll
  for (int r = 0; r < 8; ++r) Y[ybase + (size_t)r * P] = acc0[r];
  ybase += (size_t)16 * P;
#pragma unroll
  for (int r = 0; r < 8; ++r) Y[ybase + (size_t)r * P] = acc1[r];
  ybase += (size_t)16 * P;
#pragma unroll
  for (int r = 0; r < 8; ++r) Y[ybase + (size_t)r * P] = acc2[r];
  ybase += (size_t)16 * P;
#pragma unroll
  for (int r = 0; r < 8; ++r) Y[ybase + (size_t)r * P] = acc3[r];
}

// Final classifier: O=7 (padded to 16), bias added, store as (b, o, n)
__global__ __launch_bounds__(256) void conv2_wmma(
    const float* __restrict__ Wp, const float* __restrict__ bias,
    const float* __restrict__ X, float* __restrict__ out) {
  const int P = 131072, Cpad = 128;
  int wave = threadIdx.x >> 5;
  int lane = threadIdx.x & 31;
  int tileN = blockIdx.x * 8 + wave;
  int n = tileN * 16 + (lane & 15);
  int mrow = (lane & 15);
  int kh = (lane >> 4) << 1;
  v8f acc = {};
  for (int k = 0; k < Cpad; k += 4) {
    v2f a, b;
    a.x = Wp[mrow * Cpad + k + kh];
    a.y = Wp[mrow * Cpad + k + kh + 1];
    b.x = X[(size_t)(k + kh) * P + n];
    b.y = X[(size_t)(k + kh + 1) * P + n];
    acc = __builtin_amdgcn_wmma_f32_16x16x4_f32(false, a, false, b,
                                                (short)0, acc, false, false);
  }
  int mbase = (lane >> 4) << 3;
  int bb = n >> 14;        // batch (16384 points per batch)
  int nn = n & 16383;
#pragma unroll
  for (int r = 0; r < 8; ++r) {
    int m = mbase + r;
    if (m < 7) out[((size_t)bb * 7 + m) * 16384 + nn] = acc[r] + bias[m];
  }
}

// ---------------------------------------------------------------------------
// Weight padding: Wp[Opad x Cpad] from W[O x C], zero fill
// ---------------------------------------------------------------------------
__global__ __launch_bounds__(256) void pad_w(const float* __restrict__ W,
                                             float* __restrict__ Wp,
                                             int O, int C, int Cpad, int tot) {
  int t = blockIdx.x * 256 + threadIdx.x;
  if (t >= tot) return;
  int o = t / Cpad, c = t % Cpad;
  Wp[t] = (o < O && c < C) ? W[o * C + c] : 0.f;
}

// ---------------------------------------------------------------------------
// BatchNorm: per-channel stats -> fused scale/shift;   apply + ReLU
// ---------------------------------------------------------------------------
__global__ __launch_bounds__(256) void bn_stats(
    const float* __restrict__ Y, int P, const float* __restrict__ g,
    const float* __restrict__ beta, float* __restrict__ scale,
    float* __restrict__ shift) {
  __shared__ float s1[256], s2[256];
  int o = blockIdx.x;
  const float* row = Y + (size_t)o * P;
  float a = 0.f, q = 0.f;
  for (int p = threadIdx.x; p < P; p += 256) {
    float v = row[p];
    a += v; q += v * v;
  }
  s1[threadIdx.x] = a; s2[threadIdx.x] = q;
  __syncthreads();
  for (int s = 128; s > 0; s >>= 1) {
    if (threadIdx.x < s) {
      s1[threadIdx.x] += s1[threadIdx.x + s];
      s2[threadIdx.x] += s2[threadIdx.x + s];
    }
    __syncthreads();
  }
  if (threadIdx.x == 0) {
    float inv = 1.f / (float)P;
    float mu = s1[0] * inv;
    float var = fmaxf(s2[0] * inv - mu * mu, 0.f);
    float r = rsqrtf(var + 1e-5f);
    float sc = g[o] * r;
    scale[o] = sc;
    shift[o] = beta[o] - mu * sc;
  }
}

__global__ __launch_bounds__(256) void bn_apply(float* __restrict__ Y,
                                                const float* __restrict__ scale,
                                                const float* __restrict__ shift,
                                                int P) {
  int o = blockIdx.y;
  int p = blockIdx.x * 256 + threadIdx.x;
  size_t i = (size_t)o * P + p;
  float v = Y[i] * scale[o] + shift[o];
  Y[i] = v > 0.f ? v : 0.f;
}

// ---------------------------------------------------------------------------
// Max-pool over ns contiguous samples per group
// ---------------------------------------------------------------------------
__global__ __launch_bounds__(256) void maxpool_k(const float* __restrict__ Y,
                                                 float* __restrict__ out,
                                                 int ns, int Pout) {
  int o = blockIdx.y;
  int idx = blockIdx.x * 256 + threadIdx.x;
  if (idx >= Pout) return;
  const float* s = Y + ((size_t)o * Pout + idx) * (size_t)ns;
  float m = s[0];
  for (int j = 1; j < ns; ++j) m = fmaxf(m, s[j]);
  out[(size_t)o * Pout + idx] = m;
}

// ---------------------------------------------------------------------------
// Grouping / concat kernels
// ---------------------------------------------------------------------------
__global__ __launch_bounds__(256) void group_sa1(const float* __restrict__ feat,
                                                 float* __restrict__ X,
                                                 float* __restrict__ new_xyz) {
  int t = blockIdx.x * 256 + threadIdx.x;       // B*1024*32 = 262144
  if (t >= 262144) return;
  int j = t & 31, i = (t >> 5) & 1023, b = t >> 15;
  const int N = 16384, P = 262144;
  unsigned s = hash3(0x51u, b, i) & (N - 1);
  unsigned nb = hash3(0x52u, b, (i << 5) | j) & (N - 1);
  const float* pc = feat + ((size_t)b * N + s) * 6;
  const float* pn = feat + ((size_t)b * N + nb) * 6;
  X[0 * P + t] = pn[0] - pc[0];
  X[1 * P + t] = pn[1] - pc[1];
  X[2 * P + t] = pn[2] - pc[2];
  X[3 * P + t] = pn[3];
  X[4 * P + t] = pn[4];
  X[5 * P + t] = pn[5];
  X[6 * P + t] = 0.f;
  X[7 * P + t] = 0.f;
  if (j == 0) {
    float* o = new_xyz + ((size_t)b * 1024 + i) * 3;
    o[0] = pc[0]; o[1] = pc[1]; o[2] = pc[2];
  }
}

__global__ __launch_bounds__(256) void group_sa2(const float* __restrict__ l1xyz,
                                                 const float* __restrict__ l1pts,
                                                 float* __restrict__ X,
                                                 float* __restrict__ new_xyz) {
  int t = blockIdx.x * 256 + threadIdx.x;       // B*256*32 = 65536
  if (t >= 65536) return;
  int j = t & 31, i = (t >> 5) & 255, b = t >> 13;
  const int P = 65536, P1 = 8192;
  unsigned s = hash3(0x61u, b, i) & 1023;
  unsigned nb = hash3(0x62u, b, (i << 5) | j) & 1023;
  const float* pc = l1xyz + ((size_t)b * 1024 + s) * 3;
  const float* pn = l1xyz + ((size_t)b * 1024 + nb) * 3;
  X[0 * P + t] = pn[0] - pc[0];
  X[1 * P + t] = pn[1] - pc[1];
  X[2 * P + t] = pn[2] - pc[2];
  int col = b * 1024 + (int)nb;
  for (int c = 0; c < 128; ++c)
    X[(size_t)(3 + c) * P + t] = l1pts[(size_t)c * P1 + col];
  X[(size_t)131 * P + t] = 0.f;
  if (j == 0) {
    float* o = new_xyz + ((size_t)b * 256 + i) * 3;
    o[0] = pc[0]; o[1] = pc[1]; o[2] = pc[2];
  }
}

__global__ __launch_bounds__(256) void group_all_sa3(const float* __restrict__ l2xyz,
                                                     const float* __restrict__ l2pts,
                                                     float* __restrict__ X) {
  int t = blockIdx.x * 256 + threadIdx.x;       // B*256 = 2048
  if (t >= 2048) return;
  const int P = 2048;
  const float* pq = l2xyz + (size_t)t * 3;
  X[0 * P + t] = pq[0];
  X[1 * P + t] = pq[1];
  X[2 * P + t] = pq[2];
  for (int c = 0; c < 256; ++c)
    X[(size_t)(3 + c) * P + t] = l2pts[(size_t)c * P + t];
  X[(size_t)259 * P + t] = 0.f;
}

__global__ __launch_bounds__(256) void concat_fp3(const float* __restrict__ l2pts,
                                                  const float* __restrict__ l3pts,
                                                  float* __restrict__ X) {
  int t = blockIdx.x * 256 + threadIdx.x;       // 512 * 2048
  if (t >= 512 * 2048) return;
  int r = t >> 11, p = t & 2047, b = p >> 8;
  X[t] = (r < 256) ? l2pts[(size_t)r * 2048 + p] : l3pts[(r - 256) * 8 + b];
}

__global__ __launch_bounds__(256) void copy_rows(float* __restrict__ dst,
                                                 const float* __restrict__ src,
                                                 int P) {
  int p = blockIdx.x * 256 + threadIdx.x;
  size_t r = blockIdx.y;
  dst[r * P + p] = src[r * P + p];
}

__global__ __launch_bounds__(256) void zero_rows(float* __restrict__ dst, int P) {
  int p = blockIdx.x * 256 + threadIdx.x;
  size_t r = blockIdx.y;
  dst[r * P + p] = 0.f;
}

__global__ __launch_bounds__(256) void transpose_feat(const float* __restrict__ feat,
                                                      float* __restrict__ X) {
  int t = blockIdx.x * 256 + threadIdx.x;       // 6 * 131072
  if (t >= 6 * 131072) return;
  int r = t >> 17, p = t & 131071;
  X[(size_t)r * 131072 + p] = feat[(size_t)p * 6 + r];
}

// ---------------------------------------------------------------------------
// 3-NN inverse-distance interpolation (feature propagation)
// ---------------------------------------------------------------------------
__global__ __launch_bounds__(256) void fp_interp(
    const float* __restrict__ xyz1, int s1, const float* __restrict__ xyz2,
    const float* __restrict__ pts2, float* __restrict__ out,
    int N1, int N2, int C2, int P, int P2) {
  int t = blockIdx.x * 256 + threadIdx.x;
  if (t >= P) return;
  int b = t / N1;
  float ax = xyz1[(size_t)t * s1 + 0];
  float ay = xyz1[(size_t)t * s1 + 1];
  float az = xyz1[(size_t)t * s1 + 2];
  const float* B2 = xyz2 + (size_t)b * N2 * 3;
  float d0 = 3.4e38f, d1 = 3.4e38f, d2 = 3.4e38f;
  int i0 = 0, i1 = 0, i2 = 0;
  for (int m = 0; m < N2; ++m) {
    float dx = ax - B2[m * 3 + 0];
    float dy = ay - B2[m * 3 + 1];
    float dz = az - B2[m * 3 + 2];
    float d = dx * dx + dy * dy + dz * dz;
    if (d < d0)      { d2 = d1; i2 = i1; d1 = d0; i1 = i0; d0 = d; i0 = m; }
    else if (d < d1) { d2 = d1; i2 = i1; d1 = d; i1 = m; }
    else if (d < d2) { d2 = d; i2 = m; }
  }
  float w0 = 1.f / fmaxf(sqrtf(fmaxf(d0, 0.f)), 1e-10f);
  float w1 = 1.f / fmaxf(sqrtf(fmaxf(d1, 0.f)), 1e-10f);
  float w2 = 1.f / fmaxf(sqrtf(fmaxf(d2, 0.f)), 1e-10f);
  float ws = w0 + w1 + w2;
  w0 /= ws; w1 /= ws; w2 /= ws;
  int c0 = b * N2 + i0, c1 = b * N2 + i1, c2 = b * N2 + i2;
  for (int c = 0; c < C2; ++c) {
    out[(size_t)c * P + t] = w0 * pts2[(size_t)c * P2 + c0] +
                             w1 * pts2[(size_t)c * P2 + c1] +
                             w2 * pts2[(size_t)c * P2 + c2];
  }
}

// ---------------------------------------------------------------------------
// Host orchestration
// ---------------------------------------------------------------------------
struct Lyr { const float *W, *b, *g, *beta; int O, C; };

extern "C" void kernel_launch(void* const* d_in, const int* in_sizes, int n_in,
                              void* d_out, int out_size, void* d_ws, size_t ws_size,
                              hipStream_t stream) {
  (void)out_size;
  const float* features = (const float*)d_in[0];

  // ---- parameter layer shapes (O, C) ----
  const int sa1s[3][2] = {{64, 6}, {64, 64}, {128, 64}};
  const int sa2s[3][2] = {{128, 131}, {128, 128}, {256, 128}};
  const int sa3s[3][2] = {{256, 259}, {512, 256}, {256, 512}};
  const int fp3s[2][2] = {{256, 512}, {256, 256}};
  const int fp2s[2][2] = {{256, 384}, {128, 256}};
  const int fp1s[3][2] = {{128, 134}, {128, 128}, {128, 128}};
  const int heads[1][2] = {{128, 128}};

  Lyr sa1[3], sa2[3], sa3[3], fp3[2], fp2[2], fp1[3], head[1];
  const float *c2W = nullptr, *c2b = nullptr;

  // Flattening-order detection: insertion order (sa1 W first: 64*6=384)
  // vs jax-sorted order (conv2 W first: 7*128=896).
  bool sorted = (n_in > 1 && in_sizes[1] == 7 * 128);
  int idx = 1;
  auto take = [&](Lyr* arr, const int (*sh)[2], int nl) {
    for (int i = 0; i < nl; ++i) {
      arr[i].O = sh[i][0]; arr[i].C = sh[i][1];
      arr[i].W = (const float*)d_in[idx++];
      arr[i].b = (const float*)d_in[idx++];
      if (sorted) {
        arr[i].beta = (const float*)d_in[idx++];
        arr[i].g = (const float*)d_in[idx++];
      } else {
        arr[i].g = (const float*)d_in[idx++];
        arr[i].beta = (const float*)d_in[idx++];
      }
    }
  };
  if (sorted) {
    c2W = (const float*)d_in[idx++]; c2b = (const float*)d_in[idx++];
    take(fp1, fp1s, 3); take(fp2, fp2s, 2); take(fp3, fp3s, 2);
    take(head, heads, 1); take(sa1, sa1s, 3); take(sa2, sa2s, 3); take(sa3, sa3s, 3);
  } else {
    take(sa1, sa1s, 3); take(sa2, sa2s, 3); take(sa3, sa3s, 3);
    take(fp3, fp3s, 2); take(fp2, fp2s, 2); take(fp1, fp1s, 3);
    take(head, heads, 1);
    c2W = (const float*)d_in[idx++]; c2b = (const float*)d_in[idx++];
  }

  // ---- workspace carving (floats) ----
  float* w = (float*)d_ws;
  auto alloc = [&](size_t n) { float* p = w; w += (n + 255) & ~(size_t)255; return p; };
  float* XBUF   = alloc(17825792);   // max concat input (fp1: 136*131072)
  float* YA     = alloc(33554432);   // ping (128*262144)
  float* YB     = alloc(33554432);   // pong
  float* l1_xyz = alloc(24576);
  float* l2_xyz = alloc(6144);
  float* l1p    = alloc(1048576);    // sa1 pooled  128 x 8192
  float* l2p    = alloc(524288);     // sa2 pooled  256 x 2048
  float* l3p    = alloc(2048);       // sa3 pooled  256 x 8
  float* l2p2   = alloc(524288);     // fp3 out     256 x 2048
  float* l1p2   = alloc(1048576);    // fp2 out     128 x 8192
  float* Wp     = alloc(262144);     // padded weights scratch
  float* scl    = alloc(512);
  float* shf    = alloc(512);
  size_t need = (size_t)(w - (float*)d_ws) * sizeof(float);
  if (need > ws_size) return;

  auto mlp_layer = [&](const Lyr& L, const float* X, float* Y, int Cpad, int P) {
    int tot = L.O * Cpad;
    pad_w<<<(tot + 255) / 256, 256, 0, stream>>>(L.W, Wp, L.O, L.C, Cpad, tot);
    wmma_gemm_f32<<<dim3(P / 128, L.O / 64), 256, 0, stream>>>(Wp, X, Y, Cpad, P);
    bn_stats<<<L.O, 256, 0, stream>>>(Y, P, L.g, L.beta, scl, shf);
    bn_apply<<<dim3(P / 256, L.O), 256, 0, stream>>>(Y, scl, shf, P);
  };

  // ---- SA1: group N=16384 -> 1024x32, MLP [64,64,128], maxpool ----
  group_sa1<<<262144 / 256, 256, 0, stream>>>(features, XBUF, l1_xyz);
  mlp_layer(sa1[0], XBUF, YA, 8, 262144);
  mlp_layer(sa1[1], YA, YB, 64, 262144);
  mlp_layer(sa1[2], YB, YA, 64, 262144);
  maxpool_k<<<dim3(32, 128), 256, 0, stream>>>(YA, l1p, 32, 8192);

  // ---- SA2: group 1024 -> 256x32, MLP [128,128,256], maxpool ----
  group_sa2<<<65536 / 256, 256, 0, stream>>>(l1_xyz, l1p, XBUF, l2_xyz);
  mlp_layer(sa2[0], XBUF, YA, 132, 65536);
  mlp_layer(sa2[1], YA, YB, 128, 65536);
  mlp_layer(sa2[2], YB, YA, 128, 65536);
  maxpool_k<<<dim3(8, 256), 256, 0, stream>>>(YA, l2p, 32, 2048);

  // ---- SA3: group-all, MLP [256,512,256], maxpool over 256 ----
  group_all_sa3<<<2048 / 256, 256, 0, stream>>>(l2_xyz, l2p, XBUF);
  mlp_layer(sa3[0], XBUF, YA, 260, 2048);
  mlp_layer(sa3[1], YA, YB, 256, 2048);
  mlp_layer(sa3[2], YB, YA, 512, 2048);
  maxpool_k<<<dim3(1, 256), 256, 0, stream>>>(YA, l3p, 256, 8);

  // ---- FP3: broadcast l3, concat with l2_points, MLP [256,256] ----
  concat_fp3<<<(512 * 2048) / 256, 256, 0, stream>>>(l2p, l3p, XBUF);
  mlp_layer(fp3[0], XBUF, YA, 512, 2048);
  mlp_layer(fp3[1], YA, YB, 256, 2048);
  copy_rows<<<dim3(2048 / 256, 256), 256, 0, stream>>>(l2p2, YB, 2048);

  // ---- FP2: 3-NN interp (1024 <- 256), concat l1_points, MLP [256,128] ----
  copy_rows<<<dim3(8192 / 256, 128), 256, 0, stream>>>(XBUF, l1p, 8192);
  fp_interp<<<8192 / 256, 256, 0, stream>>>(l1_xyz, 3, l2_xyz, l2p2,
                                            XBUF + (size_t)128 * 8192,
                                            1024, 256, 256, 8192, 2048);
  mlp_layer(fp2[0], XBUF, YA, 384, 8192);
  mlp_layer(fp2[1], YA, YB, 256, 8192);
  copy_rows<<<dim3(8192 / 256, 128), 256, 0, stream>>>(l1p2, YB, 8192);

  // ---- FP1: 3-NN interp (16384 <- 1024), concat raw feats, MLP [128,128,128] ----
  transpose_feat<<<(6 * 131072) / 256, 256, 0, stream>>>(features, XBUF);
  fp_interp<<<131072 / 256, 256, 0, stream>>>(features, 6, l1_xyz, l1p2,
                                              XBUF + (size_t)6 * 131072,
                                              16384, 1024, 128, 131072, 8192);
  zero_rows<<<dim3(131072 / 256, 2), 256, 0, stream>>>(XBUF + (size_t)134 * 131072, 131072);
  mlp_layer(fp1[0], XBUF, YA, 136, 131072);
  mlp_layer(fp1[1], YA, YB, 128, 131072);
  mlp_layer(fp1[2], YB, YA, 128, 131072);

  // ---- Head MLP [128] ----
  mlp_layer(head[0], YA, YB, 128, 131072);

  // ---- Final classifier conv2 (7 classes) -> d_out (b, o, n) ----
  {
    int tot = 16 * 128;
    pad_w<<<(tot + 255) / 256, 256, 0, stream>>>(c2W, Wp, 7, 128, 128, tot);
    conv2_wmma<<<131072 / 128, 256, 0, stream>>>(Wp, c2b, YB, (float*)d_out);
  }
}